// DeformableTransformerEncoderLayer_60516089200683
// MI455X (gfx1250) — compile-verified
//
#include <hip/hip_runtime.h>
#include <math.h>

#define NHH 8
#define HDD 32
#define NLV 4
#define NPP 4
#define DM  256
#define BATCH 2
#define LTOK 19560
#define MTOK (BATCH * LTOK)   /* 39120, divisible by 16 */

typedef __attribute__((ext_vector_type(2))) float v2f;
typedef __attribute__((ext_vector_type(8))) float v8f;

// ---------------------------------------------------------------------------
// Weight pre-pack: Bp[(kp*N + col)*2 + {0,1}] = Bw[(2kp)*N + col], Bw[(2kp+1)*N + col]
// One float2 per (K-pair, column) -> B fragment = single global_load_b64.
// ---------------------------------------------------------------------------
__global__ __launch_bounds__(256) void pack_b_kernel(
    const float* __restrict__ Bw, float* __restrict__ Bp, int N, int K)
{
    const int i = blockIdx.x * blockDim.x + threadIdx.x;   // kp*N + col
    const int total = (K / 2) * N;
    if (i >= total) return;
    const int kp  = i / N;
    const int col = i - kp * N;
    Bp[2 * i + 0] = Bw[(size_t)(2 * kp + 0) * N + col];
    Bp[2 * i + 1] = Bw[(size_t)(2 * kp + 1) * N + col];
}

// ---------------------------------------------------------------------------
// GEMM: Dst[M,N] = (A (+ A2 if ADD2))[M,K] * B[K,N] + bias, optional ReLU.
// B is pre-packed (see pack_b_kernel). Wave tile = 16(M) x 128(N): 8 v8f
// accumulators; B fragments register-double-buffered (prefetch next k-step
// before issuing current WMMAs) so s_wait covers a full iteration of slack.
// A tile (16x256 chunk) staged in LDS, row stride 260 to spread banks.
// M % 16 == 0 (39120 = 2445*16), K % 256 == 0, N % 128 == 0.
// ---------------------------------------------------------------------------
template <bool ADD2, bool RELU>
__global__ __launch_bounds__(256) void gemm_wmma_f32(
    const float* __restrict__ A, const float* __restrict__ A2,
    const float* __restrict__ Bp, const float* __restrict__ bias,
    float* __restrict__ Dst, int M, int N, int K)
{
    constexpr int KC  = 256;
    constexpr int LDA = KC + 4;
    constexpr int NT  = 8;            // 8 x 16-col tiles per wave = 128 cols
    __shared__ float As[16 * LDA];

    const int wave = threadIdx.x >> 5;
    const int lane = threadIdx.x & 31;
    const int hi   = lane >> 4;       // 0: K pair {0,1}; 1: K pair {2,3}
    const int ln   = lane & 15;       // A: row M, B/D: column N
    const int row0 = blockIdx.x * 16;
    const int col0 = wave * (NT * 16);

    v8f acc[NT] = {};

    for (int kc = 0; kc < K; kc += KC) {
        // Stage A[row0:row0+16, kc:kc+256] into LDS (float4 granularity).
        for (int i = threadIdx.x; i < 16 * (KC / 4); i += blockDim.x) {
            const int r  = i >> 6;              // 64 float4 per row
            const int c4 = (i & 63) << 2;
            const size_t g = (size_t)(row0 + r) * K + kc + c4;
            float4 v = *reinterpret_cast<const float4*>(A + g);
            if (ADD2) {
                const float4 b = *reinterpret_cast<const float4*>(A2 + g);
                v.x += b.x; v.y += b.y; v.z += b.z; v.w += b.w;
            }
            *reinterpret_cast<float4*>(&As[r * LDA + c4]) = v;
        }
        __syncthreads();

        // Packed-B base for this chunk: kp = kc/2 + k/2 + hi, col = col0+ln.
        const float* __restrict__ bb =
            Bp + ((size_t)(kc / 2 + hi) * N + col0 + ln) * 2;

        // Preload k-step 0.
        v2f bcur[NT];
        #pragma unroll
        for (int t = 0; t < NT; ++t)
            bcur[t] = *reinterpret_cast<const v2f*>(bb + t * 32);

        #pragma unroll 4
        for (int k = 0; k < KC - 4; k += 4) {
            // Prefetch k-step (k+4) B fragments before current WMMAs.
            v2f bnxt[NT];
            const float* __restrict__ bn = bb + (size_t)(k / 2 + 2) * 2 * N;
            #pragma unroll
            for (int t = 0; t < NT; ++t)
                bnxt[t] = *reinterpret_cast<const v2f*>(bn + t * 32);

            const v2f af = *reinterpret_cast<const v2f*>(&As[ln * LDA + k + 2 * hi]);
            #pragma unroll
            for (int t = 0; t < NT; ++t)
                acc[t] = __builtin_amdgcn_wmma_f32_16x16x4_f32(
                    false, af, false, bcur[t], (short)0, acc[t], false, false);

            #pragma unroll
            for (int t = 0; t < NT; ++t) bcur[t] = bnxt[t];
        }
        { // tail k-step (k = KC-4)
            const v2f af =
                *reinterpret_cast<const v2f*>(&As[ln * LDA + (KC - 4) + 2 * hi]);
            #pragma unroll
            for (int t = 0; t < NT; ++t)
                acc[t] = __builtin_amdgcn_wmma_f32_16x16x4_f32(
                    false, af, false, bcur[t], (short)0, acc[t], false, false);
        }
        __syncthreads();
    }

    // Epilogue: VGPR r holds rows {r, 8+r} (lanes 0-15 / 16-31), N = col.
    #pragma unroll
    for (int t = 0; t < NT; ++t) {
        const int col = col0 + t * 16 + ln;
        const float bb2 = bias[col];
        #pragma unroll
        for (int r = 0; r < 8; ++r) {
            float v = acc[t][r] + bb2;
            if (RELU) v = fmaxf(v, 0.0f);
            Dst[(size_t)(row0 + r + 8 * hi) * N + col] = v;
        }
    }
}

// ---------------------------------------------------------------------------
// Softmax over the 16 (NL*NP) logits of each (token, head).
// ---------------------------------------------------------------------------
__global__ __launch_bounds__(256) void softmax16_kernel(float* __restrict__ attn, int total)
{
    const int t = blockIdx.x * blockDim.x + threadIdx.x;
    if (t >= total) return;
    float v[16];
    float m = -1e30f;
    #pragma unroll
    for (int i = 0; i < 16; ++i) { v[i] = attn[t * 16 + i]; m = fmaxf(m, v[i]); }
    float s = 0.0f;
    #pragma unroll
    for (int i = 0; i < 16; ++i) { v[i] = expf(v[i] - m); s += v[i]; }
    const float inv = 1.0f / s;
    #pragma unroll
    for (int i = 0; i < 16; ++i) attn[t * 16 + i] = v[i] * inv;
}

// ---------------------------------------------------------------------------
// Deformable bilinear sampling. One thread per (b, q, h, c); lane == channel,
// so ref/offset/attn loads are wave-uniform and value gathers coalesce.
// value: [B*L, NH*HD] (GEMM layout == [B,L,NH,HD]); out: [B,L,NH,HD].
// ---------------------------------------------------------------------------
__global__ __launch_bounds__(256) void deform_kernel(
    const float* __restrict__ value, const float* __restrict__ offs,
    const float* __restrict__ attnw, const float* __restrict__ refp,
    float* __restrict__ out)
{
    const int t   = blockIdx.x * blockDim.x + threadIdx.x;  // < MTOK*256
    const int c   = t & (HDD - 1);
    const int h   = (t >> 5) & (NHH - 1);
    const int tok = t >> 8;                                 // b*L + q
    const int b   = tok / LTOK;

    const int Hs[4] = {92, 46, 23, 12};
    const int Ws[4] = {160, 80, 40, 20};
    const int St[4] = {0, 14720, 18400, 19320};

    float acc = 0.0f;
    #pragma unroll
    for (int l = 0; l < NLV; ++l) {
        const int H = Hs[l], W = Ws[l];
        const int vrow0 = b * LTOK + St[l];
        const float rx = refp[(tok * NLV + l) * 2 + 0];
        const float ry = refp[(tok * NLV + l) * 2 + 1];
        #pragma unroll
        for (int p = 0; p < NPP; ++p) {
            const int ob = (((tok * NHH + h) * NLV + l) * NPP + p) * 2;
            const float ox = offs[ob], oy = offs[ob + 1];
            const float aw = attnw[(tok * NHH + h) * 16 + l * NPP + p];
            // loc*W - 0.5 with loc = ref + off/W  ==>  ref*W + off - 0.5
            const float x = rx * (float)W + ox - 0.5f;
            const float y = ry * (float)H + oy - 0.5f;
            const float x0f = floorf(x), y0f = floorf(y);
            const float fx = x - x0f, fy = y - y0f;
            const int x0 = (int)x0f, y0 = (int)y0f;
            float sv = 0.0f;
            #pragma unroll
            for (int dy = 0; dy < 2; ++dy) {
                #pragma unroll
                for (int dx = 0; dx < 2; ++dx) {
                    const int ix = x0 + dx, iy = y0 + dy;
                    if (ix >= 0 && ix < W && iy >= 0 && iy < H) {
                        const float w = (dx ? fx : 1.0f - fx) * (dy ? fy : 1.0f - fy);
                        const float vv =
                            value[(size_t)((vrow0 + iy * W + ix) * NHH + h) * HDD + c];
                        sv += w * vv;
                    }
                }
            }
            acc += aw * sv;
        }
    }
    out[t] = acc;
}

// ---------------------------------------------------------------------------
// Adaptive-LN weight precompute: wc/bc[b,d] = sum_g pred[b,g]*gw/gb[g,d].
// out layout: [w1c | b1c | w2c | b2c], each BATCH*DM floats.
// ---------------------------------------------------------------------------
__global__ __launch_bounds__(256) void group_consts_kernel(
    const float* __restrict__ pred,
    const float* __restrict__ gw1, const float* __restrict__ gb1,
    const float* __restrict__ gw2, const float* __restrict__ gb2,
    float* __restrict__ outc)
{
    const int b = blockIdx.x, d = threadIdx.x;
    float w1 = 0.f, bb1 = 0.f, w2 = 0.f, bb2 = 0.f;
    #pragma unroll
    for (int g = 0; g < 6; ++g) {
        const float p = pred[b * 6 + g];
        w1  += p * gw1[g * DM + d];
        bb1 += p * gb1[g * DM + d];
        w2  += p * gw2[g * DM + d];
        bb2 += p * gb2[g * DM + d];
    }
    outc[(0 * BATCH + b) * DM + d] = w1;
    outc[(1 * BATCH + b) * DM + d] = bb1;
    outc[(2 * BATCH + b) * DM + d] = w2;
    outc[(3 * BATCH + b) * DM + d] = bb2;
}

// ---------------------------------------------------------------------------
// Residual add + adaptive LayerNorm. One block per token (256 threads = D).
// ---------------------------------------------------------------------------
__global__ __launch_bounds__(256) void resid_aln_kernel(
    const float* __restrict__ a, const float* __restrict__ r,
    const float* __restrict__ wc, const float* __restrict__ bc,
    float* __restrict__ out)
{
    const int tok = blockIdx.x;
    const int d   = threadIdx.x;
    const int b   = tok / LTOK;
    const size_t idx = (size_t)tok * DM + d;
    const float x = a[idx] + r[idx];

    __shared__ float red[DM];
    red[d] = x;
    __syncthreads();
    #pragma unroll
    for (int s = DM / 2; s > 0; s >>= 1) {
        if (d < s) red[d] += red[d + s];
        __syncthreads();
    }
    const float mu = red[0] * (1.0f / DM);
    __syncthreads();
    const float dx = x - mu;
    red[d] = dx * dx;
    __syncthreads();
    #pragma unroll
    for (int s = DM / 2; s > 0; s >>= 1) {
        if (d < s) red[d] += red[d + s];
        __syncthreads();
    }
    const float var = red[0] * (1.0f / DM);
    const float xn  = dx * rsqrtf(var + 1e-5f);
    out[idx] = wc[b * DM + d] * xn + bc[b * DM + d];
}

// ---------------------------------------------------------------------------
extern "C" void kernel_launch(void* const* d_in, const int* in_sizes, int n_in,
                              void* d_out, int out_size, void* d_ws, size_t ws_size,
                              hipStream_t stream)
{
    (void)in_sizes; (void)n_in; (void)out_size; (void)ws_size;

    const float* src  = (const float*)d_in[0];
    const float* pos  = (const float*)d_in[1];
    const float* refp = (const float*)d_in[2];
    const float* pred = (const float*)d_in[3];
    // d_in[4]/d_in[5]: spatial_shapes / level_start_index (static, hardcoded)
    const float* Wv   = (const float*)d_in[6];  const float* bv   = (const float*)d_in[7];
    const float* Woff = (const float*)d_in[8];  const float* boff = (const float*)d_in[9];
    const float* Wa   = (const float*)d_in[10]; const float* ba   = (const float*)d_in[11];
    const float* Wo   = (const float*)d_in[12]; const float* bo   = (const float*)d_in[13];
    const float* W1   = (const float*)d_in[14]; const float* b1   = (const float*)d_in[15];
    const float* W2   = (const float*)d_in[16]; const float* b2   = (const float*)d_in[17];
    const float* gw1  = (const float*)d_in[18]; const float* gb1  = (const float*)d_in[19];
    const float* gw2  = (const float*)d_in[20]; const float* gb2  = (const float*)d_in[21];

    // Workspace layout (bytes); aliasing is safe per the dependency order.
    char*  ws      = (char*)d_ws;
    float* value   = (float*)(ws);                 // 40,058,880  (later reused: attn_out)
    float* offs    = (float*)(ws + 40058880);      // 40,058,880  (later reused: x1)
    float* attn    = (float*)(ws + 80117760);      // 20,029,440
    float* samp    = (float*)(ws + 100147200);     // 40,058,880  (later reused: ffn_out)
    float* hbuf    = (float*)(ws + 140206080);     // 160,235,520
    float* gcons   = (float*)(ws + 300441600);     // 8,192
    float* wv_p    = (float*)(ws + 300449792);     // 262,144
    float* woff_p  = (float*)(ws + 300711936);     // 262,144
    float* wa_p    = (float*)(ws + 300974080);     // 131,072
    float* wo_p    = (float*)(ws + 301105152);     // 262,144
    float* w1_p    = (float*)(ws + 301367296);     // 1,048,576
    float* w2_p    = (float*)(ws + 302415872);     // 1,048,576
    float* attnout = value;
    float* x1      = offs;
    float* ffnout  = samp;
    float* w1c = gcons + 0 * BATCH * DM;
    float* b1c = gcons + 1 * BATCH * DM;
    float* w2c = gcons + 2 * BATCH * DM;
    float* b2c = gcons + 3 * BATCH * DM;
    float* outp = (float*)d_out;

    // Pre-pack all weights (tiny; L2-resident afterwards).
    pack_b_kernel<<<(128 * 256 + 255) / 256, 256, 0, stream>>>(Wv,   wv_p,   256, 256);
    pack_b_kernel<<<(128 * 256 + 255) / 256, 256, 0, stream>>>(Woff, woff_p, 256, 256);
    pack_b_kernel<<<(128 * 128 + 255) / 256, 256, 0, stream>>>(Wa,   wa_p,   128, 256);
    pack_b_kernel<<<(128 * 256 + 255) / 256, 256, 0, stream>>>(Wo,   wo_p,   256, 256);
    pack_b_kernel<<<(128 * 1024 + 255) / 256, 256, 0, stream>>>(W1,  w1_p,  1024, 256);
    pack_b_kernel<<<(512 * 256 + 255) / 256, 256, 0, stream>>>(W2,   w2_p,   256, 1024);

    group_consts_kernel<<<BATCH, DM, 0, stream>>>(pred, gw1, gb1, gw2, gb2, gcons);

    const int mblk = MTOK / 16;  // 2445
    // value = src @ Wv + bv
    gemm_wmma_f32<false, false><<<mblk, 64,  0, stream>>>(src, nullptr, wv_p, bv, value, MTOK, 256, 256);
    // offsets = (src+pos) @ Woff + boff
    gemm_wmma_f32<true,  false><<<mblk, 64,  0, stream>>>(src, pos, woff_p, boff, offs, MTOK, 256, 256);
    // attn logits = (src+pos) @ Wa + ba
    gemm_wmma_f32<true,  false><<<mblk, 32,  0, stream>>>(src, pos, wa_p, ba, attn, MTOK, 128, 256);
    softmax16_kernel<<<(MTOK * NHH + 255) / 256, 256, 0, stream>>>(attn, MTOK * NHH);
    // bilinear sampling -> samp [B,L,NH,HD]
    deform_kernel<<<MTOK, 256, 0, stream>>>(value, offs, attn, refp, samp);
    // attn_out = samp @ Wo + bo   (reuses value buffer)
    gemm_wmma_f32<false, false><<<mblk, 64,  0, stream>>>(samp, nullptr, wo_p, bo, attnout, MTOK, 256, 256);
    // x1 = ALN1(src + attn_out)   (reuses offs buffer)
    resid_aln_kernel<<<MTOK, DM, 0, stream>>>(src, attnout, w1c, b1c, x1);
    // h = relu(x1 @ W1 + b1)
    gemm_wmma_f32<false, true ><<<mblk, 256, 0, stream>>>(x1, nullptr, w1_p, b1, hbuf, MTOK, 1024, 256);
    // ffn_out = h @ W2 + b2       (reuses samp buffer)
    gemm_wmma_f32<false, false><<<mblk, 64,  0, stream>>>(hbuf, nullptr, w2_p, b2, ffnout, MTOK, 256, 1024);
    // out = ALN2(x1 + ffn_out)
    resid_aln_kernel<<<MTOK, DM, 0, stream>>>(x1, ffnout, w2c, b2c, outp);
}